// decoder_14688788152697
// MI455X (gfx1250) — compile-verified
//
#include <hip/hip_runtime.h>
#include <math.h>

#define EMBD 300
#define HD   1024
#define VD   50004
#define BD   64
#define LD   1024
#define NCHUNK 16          // L / 64
#define CHL    64          // l-rows per attention chunk

typedef __attribute__((ext_vector_type(2))) float v2f;
typedef __attribute__((ext_vector_type(8))) float v8f;

__device__ __forceinline__ float sigmoidf_(float x) {
    return 1.0f / (1.0f + __expf(-x));
}

__device__ __forceinline__ float waveReduceSum(float v) {
    #pragma unroll
    for (int off = 16; off; off >>= 1) v += __shfl_xor(v, off, 32);
    return v;
}
__device__ __forceinline__ float waveReduceMax(float v) {
    #pragma unroll
    for (int off = 16; off; off >>= 1) v = fmaxf(v, __shfl_xor(v, off, 32));
    return v;
}

// ---------------------------------------------------------------------------
// C[64, N] = A[64, K] * W[N, K]^T + bias[N]   (EPI==1: tanh epilogue)
// One wave computes one 16x16 C tile via V_WMMA_F32_16X16X4_F32 over K.
// Block = 256 threads = 8 waves covering a 16 x 128 C slab.
// Ragged N: clamp the B row (only pollutes this lane's own masked column),
// mask the stores. EXEC stays all-1s through every WMMA.
// ---------------------------------------------------------------------------
template <int EPI>
__global__ __launch_bounds__(256)
void wmma_gemm_bias(const float* __restrict__ A, const float* __restrict__ W,
                    const float* __restrict__ bias, float* __restrict__ C,
                    int N, int K) {
    const int lane  = threadIdx.x & 31;
    const int wave  = threadIdx.x >> 5;
    const int mBase = (int)blockIdx.y << 4;
    const int nBase = ((int)blockIdx.x << 7) + (wave << 4);
    if (nBase >= N) return;                 // wave-uniform exit

    const int  half  = lane >> 4;
    const int  lrow  = lane & 15;
    const int  aRow  = mBase + lrow;        // M = 64, always in range
    const int  nCol  = nBase + lrow;
    const bool colOK = (nCol < N);
    const int  nColc = colOK ? nCol : (N - 1);

    const float* aPtr = A + (size_t)aRow  * K + (half << 1);
    const float* wPtr = W + (size_t)nColc * K + (half << 1);

    v8f acc = {0.f, 0.f, 0.f, 0.f, 0.f, 0.f, 0.f, 0.f};
    for (int k = 0; k < K; k += 4) {
        if ((k & 127) == 0)                 // scalar-uniform branch
            __builtin_prefetch(wPtr + k + 512, 0, 3);   // global_prefetch_b8
        v2f a = *(const v2f*)(aPtr + k);
        v2f b = *(const v2f*)(wPtr + k);
        acc = __builtin_amdgcn_wmma_f32_16x16x4_f32(
            /*neg_a=*/false, a, /*neg_b=*/false, b,
            /*c_mod=*/(short)0, acc, /*reuse_a=*/false, /*reuse_b=*/false);
    }

    const float bv = bias[nColc];
    #pragma unroll
    for (int i = 0; i < 8; ++i) {
        float v = acc[i] + bv;
        if (EPI == 1) v = tanhf(v);
        if (colOK)
            C[(size_t)(mBase + (half << 3) + i) * N + nCol] = v;
    }
}

// ---------------------------------------------------------------------------
// x_emb[b, :] = emb[input[b], :]
// ---------------------------------------------------------------------------
__global__ __launch_bounds__(320)
void gather_emb(const int* __restrict__ input, const float* __restrict__ emb,
                float* __restrict__ x) {
    const int b = blockIdx.x;
    const int t = threadIdx.x;
    if (t < EMBD)
        x[(size_t)b * EMBD + t] = emb[(size_t)input[b] * EMBD + t];
}

// ---------------------------------------------------------------------------
// GRU gates (PyTorch order r,z,n). gi/gh: [64, 3H] with biases folded in.
// h_new written to d_out hidden slot and to cat[:, H:2H].
// ---------------------------------------------------------------------------
__global__ __launch_bounds__(256)
void gru_gates(const float* __restrict__ gi, const float* __restrict__ gh,
               const float* __restrict__ h, float* __restrict__ hnew,
               float* __restrict__ cat) {
    const int idx = (int)blockIdx.x * 256 + threadIdx.x;   // B*H = 65536
    const int b = idx >> 10;
    const int n = idx & (HD - 1);
    const float* gib = gi + (size_t)b * 3 * HD;
    const float* ghb = gh + (size_t)b * 3 * HD;
    const float r  = sigmoidf_(gib[n]          + ghb[n]);
    const float z  = sigmoidf_(gib[n + HD]     + ghb[n + HD]);
    const float nn = tanhf   (gib[n + 2 * HD] + r * ghb[n + 2 * HD]);
    const float out = (1.0f - z) * nn + z * h[idx];
    hnew[idx] = out;
    cat[(size_t)b * (2 * HD) + HD + n] = out;
}

// ---------------------------------------------------------------------------
// Fused flash-style attention chunk. Block = (chunk j, batch b), 256 threads.
//   1) stage Wy[b,:] in LDS; 8 waves compute the 64 masked scores of this
//      chunk (float4 dots, wave reduce), saved to LDS + raw-logit buffer.
//   2) chunk-local m_j = max, p_l = exp(x_l - m_j), s_j = sum p_l.
//   3) partial context c_j[h] = sum_l p_l * src[l,b,h]  -- rereads this
//      chunk's 256 KB while still L2/WGP$-hot, so HBM sees src ONCE.
// ---------------------------------------------------------------------------
__global__ __launch_bounds__(256)
void attn_fused(const float* __restrict__ src, const float* __restrict__ Wy,
                const unsigned char* __restrict__ mask,
                float* __restrict__ xWy, float* __restrict__ partC,
                float* __restrict__ partM, float* __restrict__ partS) {
    __shared__ float s_wy[HD];
    __shared__ float s_x[CHL];
    __shared__ float s_p[CHL];
    const int b     = blockIdx.y;
    const int chunk = blockIdx.x;
    const int tid   = threadIdx.x;
    const int lane  = tid & 31;
    const int wave  = tid >> 5;
    const int l0    = chunk * CHL;

    for (int i = tid; i < HD; i += 256)
        s_wy[i] = Wy[(size_t)b * HD + i];
    __syncthreads();

    // phase 1: scores (8 rows per wave)
    for (int r = 0; r < 8; ++r) {
        const int ll = wave * 8 + r;                 // 0..63
        const int l  = l0 + ll;
        const float4* s4 = (const float4*)(src + ((size_t)l * BD + b) * HD);
        const float4* w4 = (const float4*)s_wy;
        float acc = 0.f;
        #pragma unroll
        for (int i = 0; i < HD / 4 / 32; ++i) {
            const float4 sv = s4[lane + i * 32];
            const float4 wv = w4[lane + i * 32];
            acc = fmaf(sv.x, wv.x, acc);
            acc = fmaf(sv.y, wv.y, acc);
            acc = fmaf(sv.z, wv.z, acc);
            acc = fmaf(sv.w, wv.w, acc);
        }
        acc = waveReduceSum(acc);
        if (lane == 0) {
            const float v = mask[(size_t)b * LD + l] ? -INFINITY : acc;
            s_x[ll] = v;
            xWy[(size_t)b * LD + l] = v;
        }
    }
    __syncthreads();

    // phase 2: chunk-local max / exp / sum
    float mj = -INFINITY;
    #pragma unroll 8
    for (int i = 0; i < CHL; ++i) mj = fmaxf(mj, s_x[i]);
    if (tid < CHL) s_p[tid] = __expf(s_x[tid] - mj);
    __syncthreads();
    if (tid == 0) {
        float sj = 0.f;
        for (int i = 0; i < CHL; ++i) sj += s_p[i];
        partM[b * NCHUNK + chunk] = mj;
        partS[b * NCHUNK + chunk] = sj;
    }

    // phase 3: partial context, h split as tid + {0,256,512,768}
    float a0 = 0.f, a1 = 0.f, a2 = 0.f, a3 = 0.f;
    const float* p = src + ((size_t)l0 * BD + b) * HD + tid;
    for (int ll = 0; ll < CHL; ++ll) {
        const float w = s_p[ll];
        const float* row = p + (size_t)ll * (BD * HD);
        a0 = fmaf(w, row[0],   a0);
        a1 = fmaf(w, row[256], a1);
        a2 = fmaf(w, row[512], a2);
        a3 = fmaf(w, row[768], a3);
    }
    float* pc = partC + (size_t)(b * NCHUNK + chunk) * HD + tid;
    pc[0] = a0; pc[256] = a1; pc[512] = a2; pc[768] = a3;
}

// ---------------------------------------------------------------------------
// Per-batch exact merge of chunk partials: M = max m_j, S = sum s_j e^{m_j-M}
// ---------------------------------------------------------------------------
__global__ __launch_bounds__(64)
void attn_ms(const float* __restrict__ partM, const float* __restrict__ partS,
             float* __restrict__ Mf, float* __restrict__ Sf) {
    const int b = threadIdx.x;      // 64 threads, 1 block
    float m = -INFINITY;
    #pragma unroll
    for (int j = 0; j < NCHUNK; ++j) m = fmaxf(m, partM[b * NCHUNK + j]);
    float s = 0.f;
    #pragma unroll
    for (int j = 0; j < NCHUNK; ++j)
        s += partS[b * NCHUNK + j] * __expf(partM[b * NCHUNK + j] - m);
    Mf[b] = m;
    Sf[b] = s;
}

// ---------------------------------------------------------------------------
// scores[b,l] = exp(xWy[b,l] - M[b]) / S[b]   (masked entries were -inf -> 0)
// ---------------------------------------------------------------------------
__global__ __launch_bounds__(256)
void attn_scores_out(const float* __restrict__ xWy, const float* __restrict__ Mf,
                     const float* __restrict__ Sf, float* __restrict__ out_sc) {
    const int b = blockIdx.x;
    const float m   = Mf[b];
    const float inv = 1.0f / Sf[b];
    for (int i = threadIdx.x; i < LD; i += 256)
        out_sc[(size_t)b * LD + i] =
            __expf(xWy[(size_t)b * LD + i] - m) * inv;
}

// ---------------------------------------------------------------------------
// context[b,h] = (1/S) * sum_j exp(m_j - M) * c_j[h]  -> cat[:, 0:H]
// ---------------------------------------------------------------------------
__global__ __launch_bounds__(256)
void attn_combine(const float* __restrict__ partC, const float* __restrict__ partM,
                  const float* __restrict__ Mf, const float* __restrict__ Sf,
                  float* __restrict__ cat) {
    const int b = blockIdx.y;
    const int h = ((int)blockIdx.x << 8) + threadIdx.x;
    const float m   = Mf[b];
    const float inv = 1.0f / Sf[b];
    float acc = 0.f;
    #pragma unroll
    for (int j = 0; j < NCHUNK; ++j) {
        const float scale = __expf(partM[b * NCHUNK + j] - m);
        acc = fmaf(scale, partC[(size_t)(b * NCHUNK + j) * HD + h], acc);
    }
    cat[(size_t)b * (2 * HD) + h] = acc * inv;
}

// ---------------------------------------------------------------------------
// In-place log_softmax over V=50004 per row. One 512-thread block per row.
// ---------------------------------------------------------------------------
__global__ __launch_bounds__(512)
void log_softmax_rows(float* __restrict__ base) {
    __shared__ float sm[16];
    const int b   = blockIdx.x;
    const int tid = threadIdx.x;
    float* row = base + (size_t)b * VD;

    float m = -INFINITY;
    for (int i = tid; i < VD; i += 512) m = fmaxf(m, row[i]);
    m = waveReduceMax(m);
    if ((tid & 31) == 0) sm[tid >> 5] = m;
    __syncthreads();
    float mAll = sm[0];
    #pragma unroll
    for (int j = 1; j < 16; ++j) mAll = fmaxf(mAll, sm[j]);
    __syncthreads();

    float s = 0.f;
    for (int i = tid; i < VD; i += 512) s += __expf(row[i] - mAll);
    s = waveReduceSum(s);
    if ((tid & 31) == 0) sm[tid >> 5] = s;
    __syncthreads();
    float sAll = 0.f;
    #pragma unroll
    for (int j = 0; j < 16; ++j) sAll += sm[j];

    const float lse = mAll + logf(sAll);
    for (int i = tid; i < VD; i += 512) row[i] = row[i] - lse;
}

// ---------------------------------------------------------------------------
extern "C" void kernel_launch(void* const* d_in, const int* in_sizes, int n_in,
                              void* d_out, int out_size, void* d_ws,
                              size_t ws_size, hipStream_t stream) {
    const int*           input  = (const int*)d_in[0];
    const float*         hidden = (const float*)d_in[1];    // [1,B,H]
    const float*         src    = (const float*)d_in[2];    // [L,B,H]
    const unsigned char* mask   = (const unsigned char*)d_in[3];
    const float*         emb    = (const float*)d_in[4];
    const float*         w_ih   = (const float*)d_in[5];    // [3H,EMB]
    const float*         w_hh   = (const float*)d_in[6];    // [3H,H]
    const float*         b_ih   = (const float*)d_in[7];
    const float*         b_hh   = (const float*)d_in[8];
    const float*         w_attn = (const float*)d_in[9];    // [H,H]
    const float*         b_attn = (const float*)d_in[10];
    const float*         w_lin  = (const float*)d_in[11];   // [H,2H]
    const float*         b_lin  = (const float*)d_in[12];
    const float*         w_out  = (const float*)d_in[13];   // [V,H]
    const float*         b_out  = (const float*)d_in[14];

    // Output: [ logp (B*V) | h_new (B*H) | scores (B*L) ]
    float* out_logp = (float*)d_out;
    float* out_hid  = out_logp + (size_t)BD * VD;
    float* out_sc   = out_hid + (size_t)BD * HD;

    // Workspace carve-up (floats; ~7.2 MB total)
    float* ws    = (float*)d_ws;
    float* x_emb = ws;                  // 64*300    = 19200
    float* gi    = x_emb + 19200;       // 64*3072   = 196608
    float* gh    = gi + 196608;         // 64*3072
    float* Wy    = gh + 196608;         // 64*1024   = 65536
    float* xWy   = Wy + 65536;          // 64*1024
    float* cat   = xWy + 65536;         // 64*2048   = 131072
    float* fused = cat + 131072;        // 64*1024
    float* partC = fused + 65536;       // 64*16*1024 = 1048576
    float* partM = partC + 1048576;     // 1024
    float* partS = partM + 1024;        // 1024
    float* Mf    = partS + 1024;        // 64
    float* Sf    = Mf + 64;             // 64

    // 1) embedding gather
    gather_emb<<<dim3(BD), dim3(320), 0, stream>>>(input, emb, x_emb);
    // 2) gi = x @ w_ih^T + b_ih        [64,3072], K=300
    wmma_gemm_bias<0><<<dim3(24, 4), 256, 0, stream>>>(x_emb, w_ih, b_ih, gi,
                                                       3 * HD, EMBD);
    // 3) gh = h @ w_hh^T + b_hh        [64,3072], K=1024
    wmma_gemm_bias<0><<<dim3(24, 4), 256, 0, stream>>>(hidden, w_hh, b_hh, gh,
                                                       3 * HD, HD);
    // 4) GRU gates -> h_new (d_out) and cat[:, H:]
    gru_gates<<<dim3((BD * HD) / 256), 256, 0, stream>>>(gi, gh, hidden,
                                                         out_hid, cat);
    // 5) Wy = h_new @ w_attn^T + b_attn   [64,1024], K=1024
    wmma_gemm_bias<0><<<dim3(8, 4), 256, 0, stream>>>(out_hid, w_attn, b_attn,
                                                      Wy, HD, HD);
    // 6) fused flash attention: scores + online-softmax partial contexts
    //    (single HBM pass over the 268 MB source_hiddens)
    attn_fused<<<dim3(NCHUNK, BD), 256, 0, stream>>>(src, Wy, mask, xWy,
                                                     partC, partM, partS);
    // 7) exact per-batch merge of chunk (m_j, s_j)
    attn_ms<<<dim3(1), 64, 0, stream>>>(partM, partS, Mf, Sf);
    // 8) normalized attention scores -> d_out
    attn_scores_out<<<dim3(BD), 256, 0, stream>>>(xWy, Mf, Sf, out_sc);
    // 9) merge partial contexts -> cat[:, :H]
    attn_combine<<<dim3(HD / 256, BD), 256, 0, stream>>>(partC, partM, Mf, Sf,
                                                         cat);
    // 10) fused = tanh(cat @ w_lin^T + b_lin)   [64,1024], K=2048
    wmma_gemm_bias<1><<<dim3(8, 4), 256, 0, stream>>>(cat, w_lin, b_lin, fused,
                                                      HD, 2 * HD);
    // 11) logits = fused @ w_out^T + b_out  [64,50004], K=1024 (dominant GEMM)
    wmma_gemm_bias<0><<<dim3((VD + 127) / 128, 4), 256, 0, stream>>>(
        fused, w_out, b_out, out_logp, VD, HD);
    // 12) in-place log_softmax over V
    log_softmax_rows<<<dim3(BD), 512, 0, stream>>>(out_logp);
}